// RoiAlign_59974923321893
// MI455X (gfx1250) — compile-verified
//
#include <hip/hip_runtime.h>
#include <stdint.h>

#define POOLN 7

// ---------------------------------------------------------------------------
// CDNA5 async global->LDS helpers (gfx1250). These lower to the
// GLOBAL_LOAD_ASYNC_TO_LDS_B128 / S_WAIT_ASYNCCNT instructions documented in
// CDNA5 ISA (tracked with ASYNCcnt, data lands directly in
// LDS without a VGPR round trip).
// ---------------------------------------------------------------------------

// Low 32 bits of a flat pointer derived from a __shared__ object are the
// wave-relative LDS byte address (flat->LDS aperture mapping uses addr[31:0]).
__device__ __forceinline__ unsigned lds_lo32(const void* p) {
  return (unsigned)(uintptr_t)p;
}

__device__ __forceinline__ void async_ld_b128(unsigned lds_byte_addr,
                                              const void* gaddr) {
  asm volatile("global_load_async_to_lds_b128 %0, %1, off"
               :
               : "v"(lds_byte_addr), "v"((uint64_t)(uintptr_t)gaddr)
               : "memory");
}

__device__ __forceinline__ void wait_async_le1() {
  asm volatile("s_wait_asynccnt 0x1" ::: "memory");
}
__device__ __forceinline__ void wait_async_le0() {
  asm volatile("s_wait_asynccnt 0x0" ::: "memory");
}

__device__ __forceinline__ int clampi(int v, int lo, int hi) {
  return v < lo ? lo : (v > hi ? hi : v);
}

// ---------------------------------------------------------------------------
// One block = one (box, py) row of 7 sample points. 256 threads = 8 wave32.
// Double-buffered LDS staging of the 4 bilinear corner rows (4 x 1KB) via
// async-to-LDS copies, pipelined one sample point ahead.
// ---------------------------------------------------------------------------
__global__ __launch_bounds__(256) void roi_align_cdna5(
    const float4* __restrict__ boxes,  // [N] (y1,x1,y2,x2)
    const float* __restrict__ feat,    // [H,W,C]
    float* __restrict__ out)           // [N,7,7,C]
{
  constexpr int H = 256, W = 256, C = 256;

  // 2 buffers x 4 corners x 256 floats = 8 KB
  __shared__ float smem[2 * 4 * C];

  const int n  = blockIdx.x;
  const int py = blockIdx.y;
  const int t  = threadIdx.x;  // 0..255

  const float4 b = boxes[n];
  const float y1 = b.x, x1 = b.y, y2 = b.z, x2 = b.w;

  // ---- per-row (py) vertical interpolation state (uniform per block) ----
  const float ty  = (float)py * (1.0f / 6.0f);
  const float yf  = (y1 * (1.0f - ty) + y2 * ty) * (float)(H - 1);
  const bool  vy  = (yf >= 0.0f) && (yf <= (float)(H - 1));
  const float y0f = floorf(yf);
  const float wy  = yf - y0f;
  const int   yi0 = clampi((int)y0f, 0, H - 1);
  const int   yi1 = clampi((int)y0f + 1, 0, H - 1);

  // ---- per-px horizontal interpolation state (7 points, unrolled) ----
  float xw[POOLN];
  int   xa[POOLN], xb[POOLN];
  bool  xv[POOLN];
#pragma unroll
  for (int px = 0; px < POOLN; ++px) {
    const float tx  = (float)px * (1.0f / 6.0f);
    const float xf  = (x1 * (1.0f - tx) + x2 * tx) * (float)(W - 1);
    xv[px]          = (xf >= 0.0f) && (xf <= (float)(W - 1));
    const float x0f = floorf(xf);
    xw[px]          = xf - x0f;
    xa[px]          = clampi((int)x0f, 0, W - 1);
    xb[px]          = clampi((int)x0f + 1, 0, W - 1);
  }

  // ---- staging role: thread t copies 16B of corner k's C-row ----
  const int k    = t >> 6;         // corner 0..3: (y0,x0) (y0,x1) (y1,x0) (y1,x1)
  const int segB = (t & 63) << 4;  // byte offset inside the 1KB corner row
  const int yk   = (k < 2) ? yi0 : yi1;

  const char* featB = (const char*)feat;
  auto stage = [&](int px, int buf) {
    const int xk = ((k & 1) == 0) ? xa[px] : xb[px];
    const char* g =
        featB + (((size_t)yk * W + (size_t)xk) * C) * sizeof(float) + segB;
    const unsigned l =
        lds_lo32((const char*)smem + (size_t)buf * (4 * C * 4) +
                 (size_t)k * (C * 4) + segB);
    async_ld_b128(l, g);
  };

  // ---- software pipeline over the 7 px points ----
  stage(0, 0);
  const size_t obase = (((size_t)n * POOLN + py) * POOLN) * C + t;

#pragma unroll
  for (int px = 0; px < POOLN; ++px) {
    if (px + 1 < POOLN) {
      stage(px + 1, (px + 1) & 1);  // prefetch next point into other buffer
      wait_async_le1();             // stage(px) done; stage(px+1) in flight
    } else {
      wait_async_le0();             // drain
    }
    __syncthreads();                // all waves' corner segments visible

    const float* bb  = smem + (px & 1) * (4 * C);
    const float  f00 = bb[0 * C + t];
    const float  f01 = bb[1 * C + t];
    const float  f10 = bb[2 * C + t];
    const float  f11 = bb[3 * C + t];

    const float wx  = xw[px];
    const float top = fmaf(f01 - f00, wx, f00);
    const float bot = fmaf(f11 - f10, wx, f10);
    float       o   = fmaf(bot - top, wy, top);
    o               = (vy && xv[px]) ? o : 0.0f;

    out[obase + (size_t)px * C] = o;

    __syncthreads();  // protect WAR on this buffer before it is re-staged
  }
}

extern "C" void kernel_launch(void* const* d_in, const int* in_sizes, int n_in,
                              void* d_out, int out_size, void* d_ws,
                              size_t ws_size, hipStream_t stream) {
  (void)n_in; (void)out_size; (void)d_ws; (void)ws_size;
  const float4* boxes = (const float4*)d_in[0];  // proposals: N x 4 f32
  const float*  feat  = (const float*)d_in[1];   // 256x256x256 f32
  float*        out   = (float*)d_out;           // N x 7 x 7 x 256 f32

  const int N = in_sizes[0] / 4;  // 1000
  dim3 grid((unsigned)N, POOLN, 1);
  roi_align_cdna5<<<grid, 256, 0, stream>>>(boxes, feat, out);
}